// MSN_60696477827769
// MI455X (gfx1250) — compile-verified
//
#include <hip/hip_runtime.h>
#include <hip/hip_bf16.h>
#include <math.h>

// ---------------------------------------------------------------------------
// MSN SoftPoolfeat forward for MI455X (gfx1250, wave32).
//
// Pipeline:
//   [memset stats]
//   gemm_bn_stats<3,4,64,false>    : y1 = w1*x + b1           (+ stats1)
//   bn_finalize                    : (sc1, sh1)
//   gemm_bn_stats<64,64,128,true>  : y2 = w2*relu(bn1(y1))+b2 (+ stats2)
//   bn_finalize                    : (sc2, sh2)
//   gemm_bn_stats<128,128,64,true> : y3 = w3*relu(bn2(y2))+b3 (+ stats3)
//   bn_finalize                    : (sc3, sh3)     [bn3 folded downstream]
//   top16_kernel                   : 16 smallest (value,index) per (b,c)
//   gather_kernel                  : feat (B,67,64,16) + sp_idx (B,3,64,16)
//
// GEMM tiles use V_WMMA_F32_16X16X4_F32 (full f32 -- the problem is HBM
// bandwidth bound: ~1 GB of traffic at 23.3 TB/s dominates 17.5 GFLOP, so
// f32 WMMA keeps precision while engaging the matrix pipes).
// N is compile-time (16384) so output-store offsets fold into instruction
// immediates (saddr-form global_store instead of 64-bit VALU address chains),
// and staging uses b128 global loads + b128 LDS stores + global_prefetch.
// ---------------------------------------------------------------------------

typedef __attribute__((ext_vector_type(2))) float v2f;
typedef __attribute__((ext_vector_type(8))) float v8f;

#define N_PTS 16384
#define NT 8   // 16-point n-tiles processed per block (amortizes stats atomics)

// D = A(16x4, weights) x B(4x16, activations) + C, K stepped by 4.
// Layouts per CDNA5 ISA 7.12.2:
//   A 16x4 f32 : lane%16 = M(row);  VGPR0 holds K = (lane<16 ? 0 : 2), VGPR1 = +1
//   B  4x16 f32: lane%16 = N(col);  VGPR0 holds K = (lane<16 ? 0 : 2), VGPR1 = +1
//   C/D 16x16  : VGPR v -> M = v + (lane<16 ? 0 : 8), N = lane%16
template <int CIN, int CIN_PAD, int COUT, bool BN_IN>
__global__ __launch_bounds__(32 * (COUT / 16)) void gemm_bn_stats(
    const float* __restrict__ in,      // (B, CIN, N)
    const float* __restrict__ w,       // (COUT, CIN) row-major
    const float* __restrict__ bias,    // (COUT)
    const float* __restrict__ s_in,    // per-channel BN scale for input (or null)
    const float* __restrict__ t_in,    // per-channel BN shift for input (or null)
    float* __restrict__ out,           // (B, COUT, N)  (pre-BN activations)
    float* __restrict__ stat_sum,      // (COUT) running sum of out
    float* __restrict__ stat_sq) {     // (COUT) running sum of out^2
  constexpr int N = N_PTS;
  constexpr int NTHREADS = 32 * (COUT / 16);
  __shared__ float4 tile4[CIN_PAD * 4];   // CIN_PAD x 16 floats, 16B aligned
  const float* tile = (const float*)tile4;

  const int tid = threadIdx.x;
  const int lane = tid & 31;
  const int mtile = tid >> 5;            // wave -> 16-output-channel tile
  const int lo = lane & 15;
  const int hf = lane >> 4;
  const int b = blockIdx.y;
  const int o = mtile * 16 + lo;

  // A fragments (weights) hoisted: CIN_PAD/2 floats per lane, zero-padded.
  v2f aF[CIN_PAD / 4];
#pragma unroll
  for (int k = 0; k < CIN_PAD / 4; ++k) {
    int c0 = 4 * k + 2 * hf;
    aF[k].x = (c0 < CIN) ? w[o * CIN + c0] : 0.0f;
    aF[k].y = (c0 + 1 < CIN) ? w[o * CIN + c0 + 1] : 0.0f;
  }
  float bias_v[8];
#pragma unroll
  for (int v = 0; v < 8; ++v) bias_v[v] = bias[mtile * 16 + v + 8 * hf];

  float ssum[8], ssq[8];
#pragma unroll
  for (int v = 0; v < 8; ++v) { ssum[v] = 0.0f; ssq[v] = 0.0f; }

  // Uniform scalar base for output stores; per-lane part fits in 32 bits.
  float* __restrict__ outb = out + ((size_t)b * COUT + mtile * 16) * (size_t)N;

  for (int it = 0; it < NT; ++it) {
    const int n0 = (blockIdx.x * NT + it) * 16;

    // Stage B tile (CIN_PAD rows x 16 points) as float4 chunks with the
    // previous layer's BN+ReLU applied on load; prefetch next n-tile.
    for (int i = tid; i < CIN_PAD * 4; i += NTHREADS) {
      int c = i >> 2, q = i & 3;
      float4 vv = make_float4(0.0f, 0.0f, 0.0f, 0.0f);
      if (c < CIN) {
        const float* p = in + ((size_t)b * CIN + c) * (size_t)N + n0 + q * 4;
        vv = *(const float4*)p;
        if (it + 1 < NT) __builtin_prefetch(p + 16, 0, 1);
        if (BN_IN) {
          float s = s_in[c], t = t_in[c];
          vv.x = fmaxf(vv.x * s + t, 0.0f);
          vv.y = fmaxf(vv.y * s + t, 0.0f);
          vv.z = fmaxf(vv.z * s + t, 0.0f);
          vv.w = fmaxf(vv.w * s + t, 0.0f);
        }
      }
      tile4[i] = vv;
    }
    __syncthreads();

    v8f acc = {};
#pragma unroll
    for (int k = 0; k < CIN_PAD / 4; ++k) {
      int c0 = 4 * k + 2 * hf;
      v2f bF;
      bF.x = tile[c0 * 16 + lo];
      bF.y = tile[(c0 + 1) * 16 + lo];
      acc = __builtin_amdgcn_wmma_f32_16x16x4_f32(false, aF[k], false, bF,
                                                  (short)0, acc, false, false);
    }

    const int voff = hf * (8 * N) + n0 + lo;   // 32-bit lane offset
#pragma unroll
    for (int v = 0; v < 8; ++v) {
      float d = acc[v] + bias_v[v];
      outb[voff + v * N] = d;                   // imm-offset saddr-form store
      ssum[v] += d;
      ssq[v] += d * d;
    }
    __syncthreads();
  }

  // Per-channel stats: reduce over the 16 lanes sharing a channel, then atomics.
#pragma unroll
  for (int v = 0; v < 8; ++v) {
#pragma unroll
    for (int m = 8; m >= 1; m >>= 1) {
      ssum[v] += __shfl_xor(ssum[v], m, 16);
      ssq[v] += __shfl_xor(ssq[v], m, 16);
    }
    if (lo == 0) {
      int oc = mtile * 16 + v + 8 * hf;
      atomicAdd(&stat_sum[oc], ssum[v]);
      atomicAdd(&stat_sq[oc], ssq[v]);
    }
  }
}

__global__ void bn_finalize(const float* __restrict__ sum,
                            const float* __restrict__ sq,
                            const float* __restrict__ g,
                            const float* __restrict__ be,
                            float* __restrict__ sc, float* __restrict__ sh,
                            int C, float invBN) {
  int c = threadIdx.x;
  if (c < C) {
    float m = sum[c] * invBN;
    float v = sq[c] * invBN - m * m;
    float s = g[c] * rsqrtf(v + 1e-5f);
    sc[c] = s;
    sh[c] = be[c] - m * s;
  }
}

// Per (b,c): indices of the 16 smallest bn3(y3[b,c,:]) in ascending order.
__global__ __launch_bounds__(256) void top16_kernel(
    const float* __restrict__ y3, const float* __restrict__ sc,
    const float* __restrict__ sh, int* __restrict__ idx_out) {
  constexpr int N = N_PTS;
  __shared__ float lv[4096];
  __shared__ int li[4096];
  const int c = blockIdx.x, b = blockIdx.y, t = threadIdx.x;
  const float s = sc[c], tt = sh[c];
  const float* row = y3 + ((size_t)b * 64 + c) * (size_t)N;

  float best[16];
  int bidx[16];
#pragma unroll
  for (int j = 0; j < 16; ++j) { best[j] = INFINITY; bidx[j] = 0x7fffffff; }

  // Register-resident sorted-16 insert (fully unrolled -> no scratch).
  for (int n = t; n < N; n += 256) {
    float bv = row[n] * s + tt;
    int bi = n;
    if (bv < best[15]) {
#pragma unroll
      for (int j = 0; j < 16; ++j) {
        bool sw = (bv < best[j]) || (bv == best[j] && bi < bidx[j]);
        float ov = best[j]; int oi = bidx[j];
        best[j] = sw ? bv : ov;  bidx[j] = sw ? bi : oi;
        bv = sw ? ov : bv;       bi = sw ? oi : bi;
      }
    }
  }
#pragma unroll
  for (int j = 0; j < 16; ++j) { lv[t * 16 + j] = best[j]; li[t * 16 + j] = bidx[j]; }
  __syncthreads();

  // One wave merges 4096 candidates: 16 rounds of tie-broken argmin.
  if (t < 32) {
    for (int r = 0; r < 16; ++r) {
      float mv = INFINITY; int mi = 0x7fffffff; int ms = 0;
      for (int sidx = t; sidx < 4096; sidx += 32) {
        float v = lv[sidx]; int i2 = li[sidx];
        if (v < mv || (v == mv && i2 < mi)) { mv = v; mi = i2; ms = sidx; }
      }
#pragma unroll
      for (int off = 16; off >= 1; off >>= 1) {
        float ov = __shfl_xor(mv, off, 32);
        int oi = __shfl_xor(mi, off, 32);
        int os = __shfl_xor(ms, off, 32);
        if (ov < mv || (ov == mv && oi < mi)) { mv = ov; mi = oi; ms = os; }
      }
      if (t == 0) {
        idx_out[((b * 64 + c) << 4) + r] = mi;
        lv[ms] = INFINITY;  // knock out winner; same-wave DS is in-order
      }
      __builtin_amdgcn_wave_barrier();
    }
  }
}

// feat[b,ch,c,n]: ch 0..2 = x gather; ch 3..66 = bn3(y3) gather.
// sp_idx written after feat as numeric floats.
__global__ __launch_bounds__(128) void gather_kernel(
    const float* __restrict__ x, const float* __restrict__ y3,
    const float* __restrict__ sc3, const float* __restrict__ sh3,
    const int* __restrict__ idxw, float* __restrict__ out, int B) {
  constexpr int N = N_PTS;
  const int bc = blockIdx.x;
  const int b = bc >> 6;
  const int c = bc & 63;
  const int t = threadIdx.x;
  __shared__ int sid[16];
  if (t < 16) sid[t] = idxw[(bc << 4) + t];
  __syncthreads();
  const size_t feat_total = (size_t)B * 67 * 64 * 16;
  for (int i = t; i < 67 * 16; i += 128) {
    int ch = i >> 4, n = i & 15;
    int id = sid[n];
    float val;
    if (ch < 3) {
      val = x[((size_t)b * 3 + ch) * (size_t)N + id];
    } else {
      int c2 = ch - 3;
      val = y3[((size_t)b * 64 + c2) * (size_t)N + id] * sc3[c2] + sh3[c2];
    }
    out[(((size_t)b * 67 + ch) * 64 + c) * 16 + n] = val;
  }
  for (int i = t; i < 3 * 16; i += 128) {
    int ch = i >> 4, n = i & 15;
    out[feat_total + (((size_t)b * 3 + ch) * 64 + c) * 16 + n] = (float)sid[n];
  }
}

extern "C" void kernel_launch(void* const* d_in, const int* in_sizes, int n_in,
                              void* d_out, int out_size, void* d_ws,
                              size_t ws_size, hipStream_t stream) {
  (void)n_in; (void)out_size; (void)ws_size;
  const int N = N_PTS;
  const int B = in_sizes[0] / (3 * N);   // 32

  const float* x  = (const float*)d_in[0];
  const float* w1 = (const float*)d_in[1];
  const float* b1 = (const float*)d_in[2];
  const float* g1 = (const float*)d_in[3];
  const float* be1 = (const float*)d_in[4];
  const float* w2 = (const float*)d_in[5];
  const float* b2 = (const float*)d_in[6];
  const float* g2 = (const float*)d_in[7];
  const float* be2 = (const float*)d_in[8];
  const float* w3 = (const float*)d_in[9];
  const float* b3 = (const float*)d_in[10];
  const float* g3 = (const float*)d_in[11];
  const float* be3 = (const float*)d_in[12];

  // Workspace layout
  char* ws = (char*)d_ws;
  size_t off = 0;
  float* y1 = (float*)(ws + off); off += (size_t)B * 64 * N * sizeof(float);
  float* y2 = (float*)(ws + off); off += (size_t)B * 128 * N * sizeof(float);
  float* y3 = (float*)(ws + off); off += (size_t)B * 64 * N * sizeof(float);
  float* stats = (float*)(ws + off); off += 512 * sizeof(float);
  float* sum1 = stats, *sq1 = stats + 64;
  float* sum2 = stats + 128, *sq2 = stats + 256;
  float* sum3 = stats + 384, *sq3 = stats + 448;
  float* sc1 = (float*)(ws + off); off += 64 * sizeof(float);
  float* sh1 = (float*)(ws + off); off += 64 * sizeof(float);
  float* sc2 = (float*)(ws + off); off += 128 * sizeof(float);
  float* sh2 = (float*)(ws + off); off += 128 * sizeof(float);
  float* sc3 = (float*)(ws + off); off += 64 * sizeof(float);
  float* sh3 = (float*)(ws + off); off += 64 * sizeof(float);
  int* idxw = (int*)(ws + off); off += (size_t)B * 64 * 16 * sizeof(int);

  // Stats must start at zero every call (graph replays don't re-poison ws).
  hipMemsetAsync(stats, 0, 512 * sizeof(float), stream);

  const float invBN = 1.0f / (float)((size_t)B * (size_t)N);
  dim3 grid(N / (16 * NT), B);   // 128 x 32 blocks

  gemm_bn_stats<3, 4, 64, false><<<grid, 128, 0, stream>>>(
      x, w1, b1, nullptr, nullptr, y1, sum1, sq1);
  bn_finalize<<<1, 128, 0, stream>>>(sum1, sq1, g1, be1, sc1, sh1, 64, invBN);

  gemm_bn_stats<64, 64, 128, true><<<grid, 256, 0, stream>>>(
      y1, w2, b2, sc1, sh1, y2, sum2, sq2);
  bn_finalize<<<1, 128, 0, stream>>>(sum2, sq2, g2, be2, sc2, sh2, 128, invBN);

  gemm_bn_stats<128, 128, 64, true><<<grid, 128, 0, stream>>>(
      y2, w3, b3, sc2, sh2, y3, sum3, sq3);
  bn_finalize<<<1, 128, 0, stream>>>(sum3, sq3, g3, be3, sc3, sh3, 64, invBN);

  top16_kernel<<<dim3(64, B), 256, 0, stream>>>(y3, sc3, sh3, idxw);
  gather_kernel<<<B * 64, 128, 0, stream>>>(x, y3, sc3, sh3, idxw,
                                            (float*)d_out, B);
}